// PairwiseSubtractionLayer_23665269801114
// MI455X (gfx1250) — compile-verified
//
#include <hip/hip_runtime.h>
#include <math.h>

#define BATCH 256
#define LEN   512
#define NF    1024

__global__ __launch_bounds__(1024, 1)
void bump_minmax_kernel(const float* __restrict__ A,   // (BATCH, LEN, 2)
                        const float* __restrict__ B,   // (2, NF)
                        float* __restrict__ out)       // (BATCH, 1)
{
    __shared__ float sA[2 * LEN];   // 4 KB: A[b] staged, layout [l*2 + c]
    __shared__ float sCol[NF];      // 4 KB: per-column mins
    __shared__ float sWave[32];

    const int tid = threadIdx.x;    // 0..1023, one thread per column f
    const int b   = blockIdx.x;

    // ---- Stage A[b] (1024 dwords) into LDS with CDNA5 async global->LDS ----
    // Each lane copies one dword; each wave issues one async op (ASYNCcnt),
    // waits on its own asynccnt, then the block barrier covers all waves.
    {
        unsigned     lds_off = (unsigned)(size_t)(&sA[tid]);          // low 32 bits = LDS offset
        const float* g       = A + (size_t)b * (2 * LEN) + tid;
        asm volatile("global_load_async_to_lds_b32 %0, %1, off"
                     :: "v"(lds_off), "v"(g) : "memory");
        asm volatile("s_wait_asynccnt 0" ::: "memory");
    }
    __syncthreads();

    // ---- Per-column min over l of t = max(|a0-b0|, |a1-b1|) ----
    const float b0 = B[tid];
    const float b1 = B[NF + tid];

    float m0 = __builtin_inff(), m1 = __builtin_inff();
    float m2 = __builtin_inff(), m3 = __builtin_inff();

    for (int l = 0; l < LEN; l += 4) {
        // 8 consecutive LDS floats -> two broadcast ds_load_b128 per iter
        const float a00 = sA[2 * l + 0], a01 = sA[2 * l + 1];
        const float a10 = sA[2 * l + 2], a11 = sA[2 * l + 3];
        const float a20 = sA[2 * l + 4], a21 = sA[2 * l + 5];
        const float a30 = sA[2 * l + 6], a31 = sA[2 * l + 7];
        m0 = fminf(m0, fmaxf(fabsf(a00 - b0), fabsf(a01 - b1)));
        m1 = fminf(m1, fmaxf(fabsf(a10 - b0), fabsf(a11 - b1)));
        m2 = fminf(m2, fmaxf(fabsf(a20 - b0), fabsf(a21 - b1)));
        m3 = fminf(m3, fmaxf(fabsf(a30 - b0), fabsf(a31 - b1)));
    }
    const float cmin = fminf(fminf(m0, m1), fminf(m2, m3));

    // ---- Pair columns (j, j+512), then block-wide max reduce ----
    sCol[tid] = cmin;
    __syncthreads();

    float r = -__builtin_inff();
    if (tid < 512) r = fminf(sCol[tid], sCol[tid + 512]);

    // wave32 shuffle max-reduce
    #pragma unroll
    for (int off = 16; off > 0; off >>= 1)
        r = fmaxf(r, __shfl_xor(r, off, 32));
    if ((tid & 31) == 0) sWave[tid >> 5] = r;
    __syncthreads();

    if (tid < 32) {
        r = sWave[tid];
        #pragma unroll
        for (int off = 16; off > 0; off >>= 1)
            r = fmaxf(r, __shfl_xor(r, off, 32));
        if (tid == 0) {
            // bump applied once per batch: sigmoid(10x) * sigmoid(-10x)
            const float x  = 10.0f * r;
            const float s1 = 1.0f / (1.0f + expf(-x));
            const float s2 = 1.0f / (1.0f + expf(x));
            out[b] = s1 * s2;
        }
    }
}

extern "C" void kernel_launch(void* const* d_in, const int* in_sizes, int n_in,
                              void* d_out, int out_size, void* d_ws, size_t ws_size,
                              hipStream_t stream)
{
    (void)in_sizes; (void)n_in; (void)d_ws; (void)ws_size; (void)out_size;
    const float* A = (const float*)d_in[0];   // (256, 512, 2) f32
    const float* B = (const float*)d_in[1];   // (2, 1024)    f32
    float*     out = (float*)d_out;           // (256, 1)     f32

    bump_minmax_kernel<<<BATCH, 1024, 0, stream>>>(A, B, out);
}